// EdgeAttrHeteroConv_27685359190070
// MI455X (gfx1250) — compile-verified
//
#include <hip/hip_runtime.h>
#include <math.h>

#define CD 128

typedef float v2f  __attribute__((ext_vector_type(2)));
typedef float v8f  __attribute__((ext_vector_type(8)));
typedef unsigned int u32x4 __attribute__((ext_vector_type(4)));
typedef int  i32x8 __attribute__((ext_vector_type(8)));
typedef int  i32x4 __attribute__((ext_vector_type(4)));

__device__ __forceinline__ float sigf(float x)  { return 1.0f / (1.0f + __expf(-x)); }
__device__ __forceinline__ float geluf(float x) { return 0.5f * x * (1.0f + erff(x * 0.7071067811865475f)); }
__device__ __forceinline__ float lreluf(float x){ return x > 0.0f ? x : 0.2f * x; }

// float atomic max via signed/unsigned int ordering trick (init must be -inf)
__device__ __forceinline__ void atomicMaxF(float* a, float v) {
  if (v >= 0.0f) atomicMax((int*)a, __float_as_int(v));
  else           atomicMin((unsigned int*)a, __float_as_uint(v));
}

// ---------------------------------------------------------------------------
// C[n,128] = A[n,128] @ W[128,128] + bias, fp32 WMMA (V_WMMA_F32_16X16X4_F32)
// block = 256 threads = 8 waves; block computes a 64-row x 128-col strip.
// A tile staged by the Tensor Data Mover into LDS with D# padding of 4 DWORDs
// per 128-DWORD row -> row stride 132 floats (bank-conflict free fragments,
// OOB rows auto-zero). Each wave keeps its whole 128x16 B tile in registers
// (32 v2f fragments) and reuses it across 4 row tiles: 128 WMMAs per wave.
// ---------------------------------------------------------------------------
__global__ __launch_bounds__(256) void gemm128_wmma(
    const float* __restrict__ A, const float* __restrict__ W,
    const float* __restrict__ bias, float* __restrict__ C, int n)
{
  __shared__ float As[64][132];
  const int tid  = threadIdx.x;
  const int lane = tid & 31;
  const int wv   = tid >> 5;
  const int rowBase = blockIdx.x * 64;

  if (wv == 0) {
    // Tensor DMA descriptor (CDNA5 ISA ch.8): 2D tile 128x64 from row-major A
    const unsigned long long ga = (unsigned long long)(A + (size_t)rowBase * CD);
    u32x4 g0;
    g0.x = 1u;                                           // count=1, user mode
    g0.y = (unsigned)(size_t)(&As[0][0]);                // lds_addr: LDS address of As
    g0.z = (unsigned)(ga & 0xFFFFFFFFu);                 // global_addr[31:0]
    g0.w = (unsigned)((ga >> 32) & 0x01FFFFFFu)          // global_addr[56:32]
         | 0x80000000u;                                  // type=2 ("image")
    i32x8 g1;
    g1[0] = (2 << 16)      // data_size = 4 bytes
          | (1 << 20)      // pad_enable
          | (6 << 22)      // pad_interval: every 128 DWORDs
          | (3 << 25);     // pad_amount: 4 DWORDs
    g1[1] = (CD & 0xFFFF) << 16;                         // tensor_dim0 = 128
    g1[2] = ((unsigned)n & 0xFFFFu) << 16;               // tensor_dim1.lo
    g1[3] = (CD << 16) | ((unsigned)n >> 16);            // tile_dim0=128 | dim1.hi
    g1[4] = 64;                                          // tile_dim1=64, tile_dim2=0
    g1[5] = CD;                                          // tensor_dim0_stride = 128
    g1[6] = 0;
    g1[7] = 0;                                           // tensor_dim1_stride unused
    i32x4 z4 = {0, 0, 0, 0};
    i32x8 z8 = {0, 0, 0, 0, 0, 0, 0, 0};
    __builtin_amdgcn_tensor_load_to_lds(g0, g1, z4, z4, z8, 0);
    __builtin_amdgcn_s_wait_tensorcnt(0);
  }
  __syncthreads();

  const int col0 = wv * 16;
  const int m    = lane & 15;          // M (A) / N (B) index
  const int kh   = (lane >> 4) * 2;    // K sub-offset per lane half

  // preload full B tile for this wave's 16 columns into registers
  v2f barr[32];
  #pragma unroll
  for (int kb = 0; kb < 32; ++kb) {
    const int k = kb * 4 + kh;
    barr[kb].x = W[(size_t)k * CD + col0 + m];
    barr[kb].y = W[(size_t)(k + 1) * CD + col0 + m];
  }

  const int col  = col0 + m;
  const float bv = bias[col];
  const int rofs = (lane >> 4) * 8;    // lanes 16-31 hold M = j + 8

  #pragma unroll
  for (int rt = 0; rt < 4; ++rt) {
    v8f acc = {0.f, 0.f, 0.f, 0.f, 0.f, 0.f, 0.f, 0.f};
    #pragma unroll
    for (int kb = 0; kb < 32; ++kb) {
      const int k = kb * 4 + kh;
      v2f a;
      a.x = As[rt * 16 + m][k];
      a.y = As[rt * 16 + m][k + 1];
      acc = __builtin_amdgcn_wmma_f32_16x16x4_f32(false, a, false, barr[kb],
                                                  (short)0, acc, false, false);
    }
    const int rbase = rowBase + rt * 16 + rofs;
    #pragma unroll
    for (int j = 0; j < 8; ++j) {
      const int r = rbase + j;
      if (r < n) C[(size_t)r * CD + col] = acc[j] + bv;
    }
  }
}

// ---------------------------------------------------------------------------
// small helpers
// ---------------------------------------------------------------------------
__global__ void fill_f32_k(float* __restrict__ p, float v, size_t n) {
  size_t i = (size_t)blockIdx.x * blockDim.x + threadIdx.x;
  if (i < n) p[i] = v;
}

__global__ void init_ms_k(float* __restrict__ m, float* __restrict__ s, int n) {
  int i = blockIdx.x * blockDim.x + threadIdx.x;
  if (i < n) { m[i] = -INFINITY; s[i] = 0.0f; }
}

// broadcast a 128-float bias row over n elements (disease output)
__global__ void fill_bias_k(float* __restrict__ out, const float* __restrict__ b, size_t n) {
  size_t i = (size_t)blockIdx.x * blockDim.x + threadIdx.x;
  if (i < n) out[i] = b[i & (CD - 1)];
}

// precompute t0 gate tables: gT[10][128] = bg + type_emb@Wg_top ; gS[8][128] = subj_emb@Wg_bot
__global__ void gate_tables_k(const float* __restrict__ te, const float* __restrict__ se,
                              const float* __restrict__ Wg, const float* __restrict__ bg,
                              float* __restrict__ gT, float* __restrict__ gS)
{
  int i = blockIdx.x * blockDim.x + threadIdx.x;
  if (i >= 18 * CD) return;
  int row = i >> 7, c = i & (CD - 1);
  if (row < 10) {
    float a = bg[c];
    for (int j = 0; j < 32; ++j) a += te[row * 32 + j] * Wg[j * CD + c];
    gT[row * CD + c] = a;
  } else {
    int r = row - 10;
    float a = 0.0f;
    for (int j = 0; j < 32; ++j) a += se[r * 32 + j] * Wg[(32 + j) * CD + c];
    gS[r * CD + c] = a;
  }
}

// ---------------------------------------------------------------------------
// gated source-message: lane holds features [4*lane, 4*lane+4)
// GATE 0: none; GATE 1: table gate; GATE 2: 3->128->128 MLP gate (W2 in LDS)
// ---------------------------------------------------------------------------
template<int GATE>
__device__ __forceinline__ float4 gated_ms(
    const float* __restrict__ hs, int sn, int e, int lane,
    const float* __restrict__ gateT, const float* __restrict__ gateS,
    const int* __restrict__ cg, const float* __restrict__ cp,
    const float* __restrict__ W1c, const float* __restrict__ b1c,
    const float* __restrict__ W2, const float* __restrict__ b2c)
{
  float4 ms = *(const float4*)(hs + (size_t)sn * CD + lane * 4);
  if (GATE == 1) {
    const int a0 = cg[2 * (size_t)e], a1 = cg[2 * (size_t)e + 1];
    const float4 gt = *(const float4*)(gateT + (size_t)a0 * CD + lane * 4);
    const float4 gs = *(const float4*)(gateS + (size_t)a1 * CD + lane * 4);
    ms.x *= sigf(gt.x + gs.x); ms.y *= sigf(gt.y + gs.y);
    ms.z *= sigf(gt.z + gs.z); ms.w *= sigf(gt.w + gs.w);
  } else if (GATE == 2) {
    const float c0 = cp[3 * (size_t)e + 0];
    const float c1 = cp[3 * (size_t)e + 1];
    const float c2 = cp[3 * (size_t)e + 2];
    const float4 w0 = *(const float4*)(W1c + 0 * CD + lane * 4);
    const float4 w1 = *(const float4*)(W1c + 1 * CD + lane * 4);
    const float4 w2 = *(const float4*)(W1c + 2 * CD + lane * 4);
    const float4 b1 = *(const float4*)(b1c + lane * 4);
    float4 g1;
    g1.x = geluf(c0 * w0.x + c1 * w1.x + c2 * w2.x + b1.x);
    g1.y = geluf(c0 * w0.y + c1 * w1.y + c2 * w2.y + b1.y);
    g1.z = geluf(c0 * w0.z + c1 * w1.z + c2 * w2.z + b1.z);
    g1.w = geluf(c0 * w0.w + c1 * w1.w + c2 * w2.w + b1.w);
    float4 a2 = *(const float4*)(b2c + lane * 4);
    #define GATE2_ACC(comp, ci)                                               \
    for (int j2 = 0; j2 < 32; ++j2) {                                         \
      const float gj = __shfl(g1.comp, j2);                                   \
      const float4 wr = *(const float4*)(W2 + (size_t)(4 * j2 + ci) * CD + lane * 4); \
      a2.x += gj * wr.x; a2.y += gj * wr.y; a2.z += gj * wr.z; a2.w += gj * wr.w; \
    }
    GATE2_ACC(x, 0) GATE2_ACC(y, 1) GATE2_ACC(z, 2) GATE2_ACC(w, 3)
    #undef GATE2_ACC
    ms.x *= sigf(a2.x); ms.y *= sigf(a2.y);
    ms.z *= sigf(a2.z); ms.w *= sigf(a2.w);
  }
  return ms;
}

// pass 1: per-edge per-head logits + segment max (one wave32 per edge)
template<int GATE>
__global__ __launch_bounds__(256) void edge_logits_k(
    const float* __restrict__ hs, const float* __restrict__ hd,
    const int* __restrict__ src, const int* __restrict__ dst,
    const float* __restrict__ attn_t,
    const float* __restrict__ gateT, const float* __restrict__ gateS,
    const int* __restrict__ cg, const float* __restrict__ cp,
    const float* __restrict__ W1c, const float* __restrict__ b1c,
    const float* __restrict__ W2c, const float* __restrict__ b2c,
    int E, float* __restrict__ logits, float* __restrict__ mbuf)
{
  __shared__ float w2s[(GATE == 2) ? CD * CD : 1];
  const float* W2 = W2c;
  if (GATE == 2) {
    for (int i = threadIdx.x; i < CD * CD; i += blockDim.x) w2s[i] = W2c[i];
    __syncthreads();
    W2 = w2s;
  }
  const int lane = threadIdx.x & 31;
  const int wib  = threadIdx.x >> 5;
  const int nw   = gridDim.x * (blockDim.x >> 5);
  const float4 at = *(const float4*)(attn_t + lane * 4);
  for (int e = blockIdx.x * (blockDim.x >> 5) + wib; e < E; e += nw) {
    const int sn = src[e], dn = dst[e];
    float4 ms = gated_ms<GATE>(hs, sn, e, lane, gateT, gateS, cg, cp, W1c, b1c, W2, b2c);
    const float4 md = *(const float4*)(hd + (size_t)dn * CD + lane * 4);
    float p = (ms.x + md.x) * at.x + (ms.y + md.y) * at.y
            + (ms.z + md.z) * at.z + (ms.w + md.w) * at.w;
    p += __shfl_xor(p, 1);
    p += __shfl_xor(p, 2);
    p += __shfl_xor(p, 4);      // lanes 8h..8h+7 now hold head-h dot
    const float lg = lreluf(p);
    if ((lane & 7) == 0) {
      const int h = lane >> 3;
      logits[(size_t)e * 4 + h] = lg;
      atomicMaxF(&mbuf[(size_t)dn * 4 + h], lg);
    }
  }
}

// pass 2: e = exp(logit - m[dst]); segment sum (one thread per edge-head)
__global__ void edge_expsum_k(const int* __restrict__ dst, float* __restrict__ lg,
                              const float* __restrict__ mbuf, float* __restrict__ sbuf, int n)
{
  int i = blockIdx.x * blockDim.x + threadIdx.x;
  if (i < n) {
    int e = i >> 2, h = i & 3;
    int d = dst[e];
    float ex = __expf(lg[i] - mbuf[(size_t)d * 4 + h]);
    lg[i] = ex;
    atomicAdd(&sbuf[(size_t)d * 4 + h], ex);
  }
}

// pass 3: acc[dst] += (e/s) * gated_ms  (one wave32 per edge)
template<int GATE>
__global__ __launch_bounds__(256) void edge_agg_k(
    const float* __restrict__ hs,
    const int* __restrict__ src, const int* __restrict__ dst,
    const float* __restrict__ gateT, const float* __restrict__ gateS,
    const int* __restrict__ cg, const float* __restrict__ cp,
    const float* __restrict__ W1c, const float* __restrict__ b1c,
    const float* __restrict__ W2c, const float* __restrict__ b2c,
    const float* __restrict__ ebuf, const float* __restrict__ sbuf,
    float* __restrict__ acc, int E)
{
  __shared__ float w2s[(GATE == 2) ? CD * CD : 1];
  const float* W2 = W2c;
  if (GATE == 2) {
    for (int i = threadIdx.x; i < CD * CD; i += blockDim.x) w2s[i] = W2c[i];
    __syncthreads();
    W2 = w2s;
  }
  const int lane = threadIdx.x & 31;
  const int wib  = threadIdx.x >> 5;
  const int nw   = gridDim.x * (blockDim.x >> 5);
  for (int e = blockIdx.x * (blockDim.x >> 5) + wib; e < E; e += nw) {
    const int sn = src[e], dn = dst[e];
    float4 ms = gated_ms<GATE>(hs, sn, e, lane, gateT, gateS, cg, cp, W1c, b1c, W2, b2c);
    const int h = lane >> 3;
    const float alpha = ebuf[(size_t)e * 4 + h] / sbuf[(size_t)dn * 4 + h];
    float* ap = acc + (size_t)dn * CD + lane * 4;
    atomicAdd(ap + 0, alpha * ms.x);
    atomicAdd(ap + 1, alpha * ms.y);
    atomicAdd(ap + 2, alpha * ms.z);
    atomicAdd(ap + 3, alpha * ms.w);
  }
}

// ---------------------------------------------------------------------------
extern "C" void kernel_launch(void* const* d_in, const int* in_sizes, int n_in,
                              void* d_out, int out_size, void* d_ws, size_t ws_size,
                              hipStream_t stream)
{
  (void)in_sizes; (void)n_in; (void)out_size; (void)ws_size;
  const float* x_chem = (const float*)d_in[0];
  const float* x_gene = (const float*)d_in[1];
  const float* x_dis  = (const float*)d_in[2];
  const float* x_path = (const float*)d_in[3];
  const int* srcs[4] = {(const int*)d_in[4], (const int*)d_in[6], (const int*)d_in[8],  (const int*)d_in[10]};
  const int* dsts[4] = {(const int*)d_in[5], (const int*)d_in[7], (const int*)d_in[9],  (const int*)d_in[11]};
  const int*   cg_attr = (const int*)d_in[12];
  const float* cp_attr = (const float*)d_in[13];
  const float* Wsrc = (const float*)d_in[14];
  const float* bsrc = (const float*)d_in[15];
  const float* Wdst = (const float*)d_in[16];
  const float* bdst = (const float*)d_in[17];
  const float* attn = (const float*)d_in[18];
  const float* type_emb = (const float*)d_in[19];
  const float* subj_emb = (const float*)d_in[20];
  const float* Wg  = (const float*)d_in[21];
  const float* bg  = (const float*)d_in[22];
  const float* W1c = (const float*)d_in[23];
  const float* b1c = (const float*)d_in[24];
  const float* W2c = (const float*)d_in[25];
  const float* b2c = (const float*)d_in[26];
  const float* Wout = (const float*)d_in[27];
  const float* bout = (const float*)d_in[28];
  float* out = (float*)d_out;

  // workspace layout (floats), total ~25.9M floats (~104 MB)
  float* ws = (float*)d_ws;
  size_t off = 0;
  float* hs_buf   = ws + off; off += (size_t)50000 * CD;
  float* hd_buf   = ws + off; off += (size_t)50000 * CD;
  float* acc_chem = ws + off; off += (size_t)50000 * CD;
  float* acc_gene = ws + off; off += (size_t)30000 * CD;
  float* acc_path = ws + off; off += (size_t)5000  * CD;
  float* logits   = ws + off; off += (size_t)500000 * 4;
  float* mbuf     = ws + off; off += (size_t)50000 * 4;
  float* sbuf     = ws + off; off += (size_t)50000 * 4;
  float* gT       = ws + off; off += (size_t)10 * CD;
  float* gS       = ws + off; off += (size_t)8  * CD;

  // zero accumulators (contiguous), precompute t0 gate tables
  const size_t accTot = (size_t)(50000 + 30000 + 5000) * CD;
  fill_f32_k<<<(unsigned)((accTot + 255) / 256), 256, 0, stream>>>(acc_chem, 0.0f, accTot);
  gate_tables_k<<<(18 * CD + 255) / 256, 256, 0, stream>>>(type_emb, subj_emb, Wg, bg, gT, gS);

  const float* xsA[4]  = {x_chem, x_gene, x_chem, x_dis};
  const float* xdA[4]  = {x_gene, x_chem, x_path, x_gene};
  const int    NsA[4]  = {50000, 30000, 50000, 10000};
  const int    NdA[4]  = {30000, 50000, 5000, 30000};
  const int    Et[4]   = {500000, 500000, 250000, 250000};
  float*       accD[4] = {acc_gene, acc_chem, acc_path, acc_gene};
  const int    gateM[4] = {1, 0, 2, 0};

  for (int t = 0; t < 4; ++t) {
    const int Ns = NsA[t], Nd = NdA[t], E = Et[t];
    gemm128_wmma<<<(Ns + 63) / 64, 256, 0, stream>>>(
        xsA[t], Wsrc + (size_t)t * CD * CD, bsrc + t * CD, hs_buf, Ns);
    gemm128_wmma<<<(Nd + 63) / 64, 256, 0, stream>>>(
        xdA[t], Wdst + (size_t)t * CD * CD, bdst + t * CD, hd_buf, Nd);
    init_ms_k<<<(Nd * 4 + 255) / 256, 256, 0, stream>>>(mbuf, sbuf, Nd * 4);

    int eb = (E + 7) / 8; if (eb > 2048) eb = 2048;
    const float* at = attn + (size_t)t * CD;
    if (gateM[t] == 0)
      edge_logits_k<0><<<eb, 256, 0, stream>>>(hs_buf, hd_buf, srcs[t], dsts[t], at,
          gT, gS, cg_attr, cp_attr, W1c, b1c, W2c, b2c, E, logits, mbuf);
    else if (gateM[t] == 1)
      edge_logits_k<1><<<eb, 256, 0, stream>>>(hs_buf, hd_buf, srcs[t], dsts[t], at,
          gT, gS, cg_attr, cp_attr, W1c, b1c, W2c, b2c, E, logits, mbuf);
    else
      edge_logits_k<2><<<eb, 256, 0, stream>>>(hs_buf, hd_buf, srcs[t], dsts[t], at,
          gT, gS, cg_attr, cp_attr, W1c, b1c, W2c, b2c, E, logits, mbuf);

    edge_expsum_k<<<(E * 4 + 255) / 256, 256, 0, stream>>>(dsts[t], logits, mbuf, sbuf, E * 4);

    if (gateM[t] == 0)
      edge_agg_k<0><<<eb, 256, 0, stream>>>(hs_buf, srcs[t], dsts[t],
          gT, gS, cg_attr, cp_attr, W1c, b1c, W2c, b2c, logits, sbuf, accD[t], E);
    else if (gateM[t] == 1)
      edge_agg_k<1><<<eb, 256, 0, stream>>>(hs_buf, srcs[t], dsts[t],
          gT, gS, cg_attr, cp_attr, W1c, b1c, W2c, b2c, logits, sbuf, accD[t], E);
    else
      edge_agg_k<2><<<eb, 256, 0, stream>>>(hs_buf, srcs[t], dsts[t],
          gT, gS, cg_attr, cp_attr, W1c, b1c, W2c, b2c, logits, sbuf, accD[t], E);
  }

  // output projections (disease never receives messages -> pure bias broadcast)
  gemm128_wmma<<<(50000 + 63) / 64, 256, 0, stream>>>(
      acc_chem, Wout + (size_t)0 * CD * CD, bout + 0 * CD, out, 50000);
  gemm128_wmma<<<(30000 + 63) / 64, 256, 0, stream>>>(
      acc_gene, Wout + (size_t)1 * CD * CD, bout + 1 * CD, out + (size_t)50000 * CD, 30000);
  fill_bias_k<<<(unsigned)(((size_t)10000 * CD + 255) / 256), 256, 0, stream>>>(
      out + (size_t)80000 * CD, bout + 2 * CD, (size_t)10000 * CD);
  gemm128_wmma<<<(5000 + 63) / 64, 256, 0, stream>>>(
      acc_path, Wout + (size_t)3 * CD * CD, bout + 3 * CD, out + (size_t)90000 * CD, 5000);
}